// RNNModel_15375982919928
// MI455X (gfx1250) — compile-verified
//
#include <hip/hip_runtime.h>

#define UNITS 128
#define VOCAB 4096
#define BATCH 128
#define SEQ   256

typedef __attribute__((ext_vector_type(2))) float v2f;
typedef __attribute__((ext_vector_type(8))) float v8f;

#define LSTR 132  // padded LDS row stride (floats) -> conflict-free 16x4 A-frag reads

// ---------------------------------------------------------------------------
// Recurrence: 8 blocks x 512 threads (16 waves). Block b owns batch rows
// [16b, 16b+16). Wave w = (gate g = w>>3, N-tile n = w&7) computes a 16x16
// tile of (h @ U_gate) per step via v_wmma_f32_16x16x4_f32, with its 128x16
// U-panel pinned in VGPRs for all 256 steps. h lives in LDS.
// ---------------------------------------------------------------------------
__global__ __launch_bounds__(512) void rnn_recurrence_kernel(
    const int*   __restrict__ inputs,
    const float* __restrict__ h0,
    const float* __restrict__ W_c,  const float* __restrict__ W_theta,
    const float* __restrict__ W_eta,
    const float* __restrict__ U_theta, const float* __restrict__ U_eta,
    const float* __restrict__ b_theta, const float* __restrict__ b_eta,
    float* __restrict__ Y, float* __restrict__ hT)
{
  __shared__ float sh_h[16][LSTR];
  __shared__ float sh_x[3][16][LSTR];     // 0 = xc, 1 = xt, 2 = xe (gathered rows)
  __shared__ float sh_gate[2][16][LSTR];  // 0 = theta, 1 = eta
  __shared__ float sh_bias[2][UNITS];
  __shared__ int   sh_tok[16];

  const int tid  = threadIdx.x;
  const int lane = tid & 31;           // wave32
  const int wave = tid >> 5;           // 0..15
  const int g    = wave >> 3;          // 0 = theta, 1 = eta
  const int n    = wave & 7;           // N tile
  const int half = lane >> 4;
  const int l16  = lane & 15;
  const int b0   = blockIdx.x * 16;

  // Pin this wave's 128x16 U-panel in registers: 32 K-fragments (2 VGPRs each).
  // B-frag layout mirrors the documented f32 A layout: VGPR0 holds K = 4s+2h,
  // VGPR1 holds K = 4s+2h+1, lanes 0-15 cols striped, halves split K by 2.
  const float* Umat = g ? U_eta : U_theta;
  const int col = n * 16 + l16;
  v2f ufrag[32];
#pragma unroll
  for (int s = 0; s < 32; ++s) {
    const int k = 4 * s + 2 * half;
    ufrag[s].x = Umat[(k + 0) * UNITS + col];
    ufrag[s].y = Umat[(k + 1) * UNITS + col];
  }

  if (tid < 2 * UNITS) {
    const float* bsrc = (tid < UNITS) ? b_theta : b_eta;
    sh_bias[tid >> 7][tid & (UNITS - 1)] = bsrc[tid & (UNITS - 1)];
  }
#pragma unroll
  for (int i = 0; i < 4; ++i) {
    const int idx = tid + i * 512;
    sh_h[idx >> 7][idx & 127] = h0[(b0 + (idx >> 7)) * UNITS + (idx & 127)];
  }
  __syncthreads();

  const float* Wtab[3] = { W_c, W_theta, W_eta };

  for (int t = 0; t < SEQ; ++t) {
    if (tid < 16) sh_tok[tid] = inputs[(b0 + tid) * SEQ + t];
    __syncthreads();

    // one-hot "matmuls" are row gathers: 3 tables x 16 rows x 128 cols
#pragma unroll
    for (int i = 0; i < 12; ++i) {
      const int idx = tid + i * 512;        // < 6144
      const int tab = idx >> 11;
      const int r   = (idx >> 7) & 15;
      const int c   = idx & 127;
      sh_x[tab][r][c] = Wtab[tab][sh_tok[r] * UNITS + c];
    }
    __syncthreads();

    // 16x128 @ 128x16 on f32 WMMA, A streamed from LDS h
    v8f acc = {};
#pragma unroll
    for (int s = 0; s < 32; ++s) {
      const v2f a = *(const v2f*)&sh_h[l16][4 * s + 2 * half];
      acc = __builtin_amdgcn_wmma_f32_16x16x4_f32(false, a, false, ufrag[s],
                                                  (short)0, acc, false, false);
    }
#pragma unroll
    for (int v = 0; v < 8; ++v) {
      const int row = v + 8 * half;
      const float z = acc[v] + sh_bias[g][col] + sh_x[1 + g][row][col];
      sh_gate[g][row][col] = 1.0f / (1.0f + __expf(-z));
    }
    __syncthreads();

    // h_new = theta * tanh(h) + eta * tanh(xc); read phase, barrier, write phase
    float hn[4];
#pragma unroll
    for (int i = 0; i < 4; ++i) {
      const int idx = tid + i * 512;
      const int r = idx >> 7, c = idx & 127;
      hn[i] = sh_gate[0][r][c] * tanhf(sh_h[r][c]) +
              sh_gate[1][r][c] * tanhf(sh_x[0][r][c]);
    }
    __syncthreads();
#pragma unroll
    for (int i = 0; i < 4; ++i) {
      const int idx = tid + i * 512;
      const int r = idx >> 7, c = idx & 127;
      sh_h[r][c] = hn[i];
      Y[((size_t)t * BATCH + b0 + r) * UNITS + c] = hn[i];
      if (t == SEQ - 1) hT[(b0 + r) * UNITS + c] = hn[i];
    }
    __syncthreads();
  }
}

// ---------------------------------------------------------------------------
// Dense head: logits[32768,4096] = Y[32768,128] @ dense_W[128,4096] + b.
// Block = 256 threads (8 waves) computes a 128x64 tile; dense_W slab staged
// in LDS (stride 65), each wave holds its 16x128 Y panel in 64 VGPRs and
// runs 4 N-tiles x 32 K-step f32 WMMAs.
// ---------------------------------------------------------------------------
__global__ __launch_bounds__(256) void rnn_dense_kernel(
    const float* __restrict__ Y, const float* __restrict__ dW,
    const float* __restrict__ db, float* __restrict__ out)
{
  __shared__ float sB[UNITS][65];

  const int tid  = threadIdx.x;
  const int lane = tid & 31;
  const int wave = tid >> 5;            // 0..7
  const int half = lane >> 4;
  const int l16  = lane & 15;
  const int n0   = blockIdx.x * 64;
  const int m0   = blockIdx.y * 128 + wave * 16;

#pragma unroll
  for (int i = 0; i < 32; ++i) {
    const int idx = tid + i * 256;      // < 8192
    sB[idx >> 6][idx & 63] = dW[(idx >> 6) * VOCAB + n0 + (idx & 63)];
  }
  __syncthreads();

  v2f areg[32];
#pragma unroll
  for (int s = 0; s < 32; ++s)
    areg[s] = *(const v2f*)&Y[(size_t)(m0 + l16) * UNITS + 4 * s + 2 * half];

#pragma unroll
  for (int nt = 0; nt < 4; ++nt) {
    const int c = nt * 16 + l16;
    const float bias = db[n0 + c];
    v8f acc = {};
#pragma unroll
    for (int s = 0; s < 32; ++s) {
      const int k = 4 * s + 2 * half;
      v2f b;
      b.x = sB[k + 0][c];
      b.y = sB[k + 1][c];
      acc = __builtin_amdgcn_wmma_f32_16x16x4_f32(false, areg[s], false, b,
                                                  (short)0, acc, false, false);
    }
#pragma unroll
    for (int v = 0; v < 8; ++v)
      out[(size_t)(m0 + v + 8 * half) * VOCAB + n0 + c] = acc[v] + bias;
  }
}

// ---------------------------------------------------------------------------
extern "C" void kernel_launch(void* const* d_in, const int* in_sizes, int n_in,
                              void* d_out, int out_size, void* d_ws, size_t ws_size,
                              hipStream_t stream) {
  const int*   inputs  = (const int*)  d_in[0];
  const float* h0      = (const float*)d_in[1];
  const float* W_c     = (const float*)d_in[2];
  const float* W_theta = (const float*)d_in[3];
  const float* W_eta   = (const float*)d_in[4];
  const float* U_theta = (const float*)d_in[5];
  const float* U_eta   = (const float*)d_in[6];
  const float* b_theta = (const float*)d_in[7];
  const float* b_eta   = (const float*)d_in[8];
  const float* dense_W = (const float*)d_in[9];
  const float* dense_b = (const float*)d_in[10];

  float* logits = (float*)d_out;                               // [T*B, V]
  float* hT     = logits + (size_t)SEQ * BATCH * VOCAB;        // [B, U]
  float* Yws    = (float*)d_ws;                                // [T*B, U] = 16 MB

  rnn_recurrence_kernel<<<BATCH / 16, 512, 0, stream>>>(
      inputs, h0, W_c, W_theta, W_eta, U_theta, U_eta, b_theta, b_eta,
      Yws, hT);

  dim3 grid(VOCAB / 64, (SEQ * BATCH) / 128);
  rnn_dense_kernel<<<grid, 256, 0, stream>>>(Yws, dense_W, dense_b, logits);
}